// ResidualBlock_1743756722504
// MI455X (gfx1250) — compile-verified
//
#include <hip/hip_runtime.h>

typedef __attribute__((ext_vector_type(2))) float v2f;
typedef __attribute__((ext_vector_type(8))) float v8f;

namespace {
constexpr int kDModel = 32;
constexpr int kHidden = 64;
constexpr unsigned kG3 = 1u << 21; // 128^3 voxels per batch sample
constexpr int kLog2G3 = 21;
constexpr int kUnroll = 4;         // tiles (of 16 voxels) per loop body
}

// Precompute v[j] = sum_c w_out[c] * W2[c][j] (64 floats) and
// cconst = w_out . b2 + b_out, into d_ws[0..64].
__global__ void resblock_prep(const float* __restrict__ w2,
                              const float* __restrict__ b2,
                              const float* __restrict__ w_out,
                              const float* __restrict__ b_out,
                              float* __restrict__ ws) {
  const int j = threadIdx.x;
  if (j < kHidden) {
    float s = 0.f;
#pragma unroll
    for (int c = 0; c < kDModel; ++c) s = fmaf(w_out[c], w2[c * kHidden + j], s);
    ws[j] = s;
  }
  if (j == 0) {
    float t = b_out[0];
#pragma unroll
    for (int c = 0; c < kDModel; ++c) t = fmaf(w_out[c], b2[c], t);
    ws[kHidden] = t;
  }
}

// One wave owns a contiguous chunk of 16-voxel tiles inside one batch sample.
// Per tile: H(64x16) = W1(64x32) @ X(32x16) via 4 Mtiles x 8 Ksteps of
// V_WMMA_F32_16X16X4_F32 (bias fed as C of the first K-step), then
// out = w_out.x + v.relu(H) + cconst in the VALU epilogue.
__global__ __launch_bounds__(256)
void resblock_main(const float* __restrict__ x,
                   const float* __restrict__ w1,
                   const float* __restrict__ b1,
                   const float* __restrict__ w_out,
                   const float* __restrict__ ws,
                   float* __restrict__ out,
                   unsigned tiles_per_wave) {
  const int lane = (int)(threadIdx.x & 31u);
  const int l16  = lane & 15;
  const bool lo  = lane < 16;
  const int off  = lo ? 0 : 2; // K offset of this half-wave in A/B operands
  const int hi8  = lo ? 0 : 8; // M offset of this half-wave in C/D operands
  const unsigned wave = (blockIdx.x << 3) + ((unsigned)threadIdx.x >> 5);

  // ---- one-time register-resident operands --------------------------------
  v2f a[4][8];   // W1 tiles as WMMA A operands
  v2f wo[8];     // w_out channels owned by this lane
  v8f bias[4];   // b1 laid out as WMMA C
  v8f vv[4];     // v = w_out@W2 laid out to match C/D rows
#pragma unroll
  for (int mt = 0; mt < 4; ++mt) {
#pragma unroll
    for (int kc = 0; kc < 8; ++kc) {
      const float* p = w1 + (mt * 16 + l16) * kDModel + kc * 4 + off;
      v2f t; t.x = p[0]; t.y = p[1];
      a[mt][kc] = t;
    }
#pragma unroll
    for (int r = 0; r < 8; ++r) {
      bias[mt][r] = b1[mt * 16 + r + hi8];
      vv[mt][r]   = ws[mt * 16 + r + hi8];
    }
  }
#pragma unroll
  for (int kc = 0; kc < 8; ++kc) {
    v2f t; t.x = w_out[kc * 4 + off]; t.y = w_out[kc * 4 + off + 1];
    wo[kc] = t;
  }
  const float cconst = ws[kHidden];

  // ---- contiguous chunk for this wave (never straddles a batch sample) ----
  const unsigned v0 = wave * tiles_per_wave * 16u; // first voxel of chunk
  const unsigned b  = v0 >> kLog2G3;
  const float* base = x + ((size_t)b * (size_t)kDModel << kLog2G3)
                        + (v0 & (kG3 - 1u)) + (unsigned)l16;
  float* outp = out + v0 + (unsigned)l16;

  for (unsigned it = 0; it < tiles_per_wave; it += kUnroll) {
    // Issue all loads for kUnroll tiles first: 8KB in flight per wave.
    v2f Bv[kUnroll][8];
#pragma unroll
    for (int u = 0; u < kUnroll; ++u) {
#pragma unroll
      for (int kc = 0; kc < 8; ++kc) {
        const float* p0 = base + ((size_t)(kc * 4 + off) << kLog2G3) + u * 16;
        v2f t;
        t.x = p0[0];
        t.y = p0[(size_t)1 << kLog2G3];
        Bv[u][kc] = t;
      }
    }

#pragma unroll
    for (int u = 0; u < kUnroll; ++u) {
      // K-step 0 takes bias directly as the C operand (no acc-init moves).
      v8f acc0 = __builtin_amdgcn_wmma_f32_16x16x4_f32(false, a[0][0], false, Bv[u][0], (short)0, bias[0], false, false);
      v8f acc1 = __builtin_amdgcn_wmma_f32_16x16x4_f32(false, a[1][0], false, Bv[u][0], (short)0, bias[1], false, false);
      v8f acc2 = __builtin_amdgcn_wmma_f32_16x16x4_f32(false, a[2][0], false, Bv[u][0], (short)0, bias[2], false, false);
      v8f acc3 = __builtin_amdgcn_wmma_f32_16x16x4_f32(false, a[3][0], false, Bv[u][0], (short)0, bias[3], false, false);
#pragma unroll
      for (int kc = 1; kc < 8; ++kc) {
        acc0 = __builtin_amdgcn_wmma_f32_16x16x4_f32(false, a[0][kc], false, Bv[u][kc], (short)0, acc0, false, false);
        acc1 = __builtin_amdgcn_wmma_f32_16x16x4_f32(false, a[1][kc], false, Bv[u][kc], (short)0, acc1, false, false);
        acc2 = __builtin_amdgcn_wmma_f32_16x16x4_f32(false, a[2][kc], false, Bv[u][kc], (short)0, acc2, false, false);
        acc3 = __builtin_amdgcn_wmma_f32_16x16x4_f32(false, a[3][kc], false, Bv[u][kc], (short)0, acc3, false, false);
      }

      // Epilogue: direct w_out.x term plus v.relu(H); this lane holds half the
      // channels/rows of column l16, lane^16 holds the other half.
      float p = 0.f;
#pragma unroll
      for (int kc = 0; kc < 8; ++kc) {
        p = fmaf(wo[kc].x, Bv[u][kc].x, p);
        p = fmaf(wo[kc].y, Bv[u][kc].y, p);
      }
#pragma unroll
      for (int r = 0; r < 8; ++r) {
        // relu in a single v_med3: med3(x, 0, +MAX) == clamp(x, 0, +MAX)
        p = fmaf(vv[0][r], __builtin_amdgcn_fmed3f(acc0[r], 0.f, __FLT_MAX__), p);
        p = fmaf(vv[1][r], __builtin_amdgcn_fmed3f(acc1[r], 0.f, __FLT_MAX__), p);
        p = fmaf(vv[2][r], __builtin_amdgcn_fmed3f(acc2[r], 0.f, __FLT_MAX__), p);
        p = fmaf(vv[3][r], __builtin_amdgcn_fmed3f(acc3[r], 0.f, __FLT_MAX__), p);
      }
      p += __shfl_xor(p, 16, 32);
      if (lo) {
        __builtin_nontemporal_store(p + cconst, outp + u * 16);
      }
    }
    base += 16u * kUnroll;
    outp += 16u * kUnroll;
  }
}

extern "C" void kernel_launch(void* const* d_in, const int* in_sizes, int n_in,
                              void* d_out, int out_size, void* d_ws, size_t ws_size,
                              hipStream_t stream) {
  const float* x     = (const float*)d_in[0];
  const float* w1    = (const float*)d_in[1];
  const float* b1    = (const float*)d_in[2];
  const float* w2    = (const float*)d_in[3];
  const float* b2    = (const float*)d_in[4];
  const float* w_out = (const float*)d_in[5];
  const float* b_out = (const float*)d_in[6];
  float* out = (float*)d_out;
  float* ws  = (float*)d_ws;

  resblock_prep<<<1, 64, 0, stream>>>(w2, b2, w_out, b_out, ws);

  const unsigned nvox   = (unsigned)(in_sizes[0] / kDModel); // B * 128^3
  const unsigned ntiles = nvox >> 4;                         // 16 voxels/tile
  const unsigned blocks = 1024;
  const unsigned nwaves = blocks * 8;                        // 8192 waves
  const unsigned tpw    = ntiles / nwaves;                   // 32 tiles/wave
  resblock_main<<<blocks, 256, 0, stream>>>(x, w1, b1, w_out, ws, out, tpw);
}